// DepthwiseTensorProductModuleDict_62148176773683
// MI455X (gfx1250) — compile-verified
//
#include <hip/hip_runtime.h>

typedef float v2f __attribute__((ext_vector_type(2)));
typedef float v8f __attribute__((ext_vector_type(8)));

#define E_TOT 131072
#define MULW  64
#define FEAW  256
#define RADW  64
#define HIDW  128
#define WNUMW 256
#define WAVES 2
#define TILE_E (WAVES * 32)   // 32 edges per wave

__global__ __launch_bounds__(64) void
dtp_mlp_fused_kernel(const float* __restrict__ fea,
                     const float* __restrict__ vec,
                     const float* __restrict__ len,
                     const float* __restrict__ W1,
                     const float* __restrict__ b1,
                     const float* __restrict__ g,
                     const float* __restrict__ be,
                     const float* __restrict__ W2,
                     float* __restrict__ out)
{
    // 2 waves x 8192 floats (32 KB) = 64 KB LDS
    __shared__ float lds[WAVES * 8192];

    const int tid  = threadIdx.x;
    const int wave = tid >> 5;
    const int lane = tid & 31;
    const int col  = lane & 15;     // N (or M for A-frag rows) within 16
    const int half = lane >> 4;     // 0: K=0,1 ; 1: K=2,3 for f32 frags
    const int kb   = half * 2;

    float* wls = lds + wave * 8192;                 // this wave's region
    const int e0 = blockIdx.x * TILE_E + wave * 32; // first edge of this wave

    // ---- stage edge_len tile (32 x 64) into LDS, coalesced ----------------
    {
        const float4* src = (const float4*)(len + (size_t)e0 * RADW);
        float4* dst = (float4*)(wls + 4096);
        #pragma unroll
        for (int i = lane; i < 512; i += 32) dst[i] = src[i];
    }
    __syncthreads();

    // per-column affine params for the 8 hidden N-tiles
    float gv[8], bev[8];
    #pragma unroll
    for (int nt = 0; nt < 8; ++nt) {
        gv[nt]  = g[nt * 16 + col];
        bev[nt] = be[nt * 16 + col];
    }

    // ---- GEMM1 + LayerNorm + SiLU, one 16-row tile at a time --------------
    const float* lenT = wls + 4096;   // 32x64 row-major
    #pragma unroll
    for (int t = 0; t < 2; ++t) {
        v2f a1[16];
        #pragma unroll
        for (int kc = 0; kc < 16; ++kc) {
            a1[kc].x = lenT[(t * 16 + col) * RADW + kc * 4 + kb];
            a1[kc].y = lenT[(t * 16 + col) * RADW + kc * 4 + kb + 1];
        }

        v8f h[8];
        #pragma unroll
        for (int nt = 0; nt < 8; ++nt) {
            const int n = nt * 16 + col;
            // batch-load all 16 B fragments first -> one long load clause
            v2f bf[16];
            #pragma unroll
            for (int kc = 0; kc < 16; ++kc) {
                const int k = kc * 4 + kb;
                bf[kc].x = W1[(size_t)k * HIDW + n];
                bf[kc].y = W1[(size_t)(k + 1) * HIDW + n];
            }
            const float bvv = b1[n];
            v8f acc = {bvv, bvv, bvv, bvv, bvv, bvv, bvv, bvv};
            #pragma unroll
            for (int kc = 0; kc < 16; ++kc)
                acc = __builtin_amdgcn_wmma_f32_16x16x4_f32(
                    false, a1[kc], false, bf[kc], (short)0, acc, false, false);
            h[nt] = acc;
        }

        // LayerNorm: VGPR slot v holds row M = v + 8*half within this tile;
        // the row spans the 16 lanes of the half-wave -> shfl_xor reduction.
        #pragma unroll
        for (int v = 0; v < 8; ++v) {
            float s = 0.f, sq = 0.f;
            #pragma unroll
            for (int nt = 0; nt < 8; ++nt) {
                const float x = h[nt][v];
                s  += x;
                sq += x * x;
            }
            #pragma unroll
            for (int off = 1; off < 16; off <<= 1) {
                s  += __shfl_xor(s,  off, 32);
                sq += __shfl_xor(sq, off, 32);
            }
            const float mu  = s * (1.0f / HIDW);
            const float var = sq * (1.0f / HIDW) - mu * mu;
            const float inv = rsqrtf(var + 1e-5f);
            const int row = t * 16 + v + 8 * half;
            #pragma unroll
            for (int nt = 0; nt < 8; ++nt) {
                const int n = nt * 16 + col;
                const float x = (h[nt][v] - mu) * inv * gv[nt] + bev[nt];
                wls[row * HIDW + n] = x / (1.0f + __expf(-x));  // SiLU
            }
        }
    }
    __syncthreads();

    // ---- preload GEMM2 A-fragments for both 16-row tiles (128 VGPRs) ------
    v2f a2[64];   // [t*32 + kc]
    #pragma unroll
    for (int t = 0; t < 2; ++t)
        #pragma unroll
        for (int kc = 0; kc < 32; ++kc) {
            a2[t * 32 + kc].x = wls[(t * 16 + col) * HIDW + kc * 4 + kb];
            a2[t * 32 + kc].y = wls[(t * 16 + col) * HIDW + kc * 4 + kb + 1];
        }
    __syncthreads();

    // ---- GEMM2: w = S @ W2 (32x256 per wave), B frags shared by 2 tiles ---
    #pragma unroll 2
    for (int nt = 0; nt < 16; ++nt) {
        const int n = nt * 16 + col;
        v8f acc0 = {0.f, 0.f, 0.f, 0.f, 0.f, 0.f, 0.f, 0.f};
        v8f acc1 = {0.f, 0.f, 0.f, 0.f, 0.f, 0.f, 0.f, 0.f};
        #pragma unroll
        for (int kh = 0; kh < 2; ++kh) {
            v2f bf[16];
            #pragma unroll
            for (int j = 0; j < 16; ++j) {
                const int k = (kh * 16 + j) * 4 + kb;
                bf[j].x = W2[(size_t)k * WNUMW + n];
                bf[j].y = W2[(size_t)(k + 1) * WNUMW + n];
            }
            #pragma unroll
            for (int j = 0; j < 16; ++j) {
                acc0 = __builtin_amdgcn_wmma_f32_16x16x4_f32(
                    false, a2[kh * 16 + j],      false, bf[j], (short)0, acc0, false, false);
                acc1 = __builtin_amdgcn_wmma_f32_16x16x4_f32(
                    false, a2[32 + kh * 16 + j], false, bf[j], (short)0, acc1, false, false);
            }
        }
        #pragma unroll
        for (int v = 0; v < 8; ++v) {
            wls[(v + 8 * half) * WNUMW + n]        = acc0[v];
            wls[(16 + v + 8 * half) * WNUMW + n]   = acc1[v];
        }
    }
    __syncthreads();

    // ---- fused DTP tail: edge index is wave-uniform -----------------------
    const float inv_s3 = 0.5773502691896258f;   // 1/sqrt(3)
    const float inv_s2 = 0.7071067811865476f;   // 1/sqrt(2)
    #pragma unroll 2
    for (int el = 0; el < 32; ++el) {
        const size_t e = (size_t)(e0 + el);
        const float y0  = vec[e * 4 + 0];
        const float y1a = vec[e * 4 + 1];
        const float y1b = vec[e * 4 + 2];
        const float y1c = vec[e * 4 + 3];
        #pragma unroll
        for (int p = 0; p < 2; ++p) {
            const int m = lane + p * 32;
            const float x0  = fea[e * FEAW + m];
            const float x1a = fea[e * FEAW + MULW + 3 * m + 0];
            const float x1b = fea[e * FEAW + MULW + 3 * m + 1];
            const float x1c = fea[e * FEAW + MULW + 3 * m + 2];

            const float w1 = wls[el * WNUMW + m];
            const float w2 = wls[el * WNUMW + 64 + m];
            const float w3 = wls[el * WNUMW + 128 + m];
            const float w4 = wls[el * WNUMW + 192 + m];

            const float dot = x1a * y1a + x1b * y1b + x1c * y1c;
            out[e * FEAW + m] = (w1 * x0 * y0 + w4 * dot * inv_s3) * inv_s2;
            out[e * FEAW + MULW + 3 * m + 0] = (w2 * x0 * y1a + w3 * x1a * y0) * inv_s2;
            out[e * FEAW + MULW + 3 * m + 1] = (w2 * x0 * y1b + w3 * x1b * y0) * inv_s2;
            out[e * FEAW + MULW + 3 * m + 2] = (w2 * x0 * y1c + w3 * x1c * y0) * inv_s2;
        }
    }
}

extern "C" void kernel_launch(void* const* d_in, const int* in_sizes, int n_in,
                              void* d_out, int out_size, void* d_ws, size_t ws_size,
                              hipStream_t stream) {
    (void)in_sizes; (void)n_in; (void)out_size; (void)d_ws; (void)ws_size;

    // setup_inputs order: [fea, vec, len, W1, b1, g, be, W2] for 'a', then 'b'
    const float* fea_a = (const float*)d_in[0];
    const float* vec_a = (const float*)d_in[1];
    const float* len_a = (const float*)d_in[2];
    const float* W1_a  = (const float*)d_in[3];
    const float* b1_a  = (const float*)d_in[4];
    const float* g_a   = (const float*)d_in[5];
    const float* be_a  = (const float*)d_in[6];
    const float* W2_a  = (const float*)d_in[7];
    const float* fea_b = (const float*)d_in[8];
    const float* vec_b = (const float*)d_in[9];
    const float* len_b = (const float*)d_in[10];
    const float* W1_b  = (const float*)d_in[11];
    const float* b1_b  = (const float*)d_in[12];
    const float* g_b   = (const float*)d_in[13];
    const float* be_b  = (const float*)d_in[14];
    const float* W2_b  = (const float*)d_in[15];

    float* out_a = (float*)d_out;
    float* out_b = out_a + (size_t)E_TOT * FEAW;

    dim3 grid(E_TOT / TILE_E);
    dim3 block(WAVES * 32);
    dtp_mlp_fused_kernel<<<grid, block, 0, stream>>>(
        fea_a, vec_a, len_a, W1_a, b1_a, g_a, be_a, W2_a, out_a);
    dtp_mlp_fused_kernel<<<grid, block, 0, stream>>>(
        fea_b, vec_b, len_b, W1_b, b1_b, g_b, be_b, W2_b, out_b);
}